// InfoNCELoss_76355928588718
// MI455X (gfx1250) — compile-verified
//
#include <hip/hip_runtime.h>
#include <hip/hip_bf16.h>
#include <math.h>

typedef float v2f __attribute__((ext_vector_type(2)));
typedef float v8f __attribute__((ext_vector_type(8)));

#define TEMPERATURE 1.5f
#define NORM_EPS    1e-12f

__global__ void infonce_zero_kernel(float* out) {
    if (threadIdx.x == 0 && blockIdx.x == 0) out[0] = 0.0f;
}

// One wave (32 lanes) owns a tile of 16 rows.
// fp32 WMMA 16x16x4 operand layout (ISA 7.12.2):
//   A (16x4): lane L<16 -> row M=L, VGPRs {K=0,K=1}; lane L+16 -> row M=L, VGPRs {K=2,K=3}
//   B (4x16): same striping with N=lane -> loading P row-major with the SAME per-lane
//   addressing yields exactly P^T as the B operand. So one b64 load per matrix per step.
// diag(acc_AP) = a.p per row, diag(acc_NA) = n.a per row. Norms accumulate on VALU.
__global__ __launch_bounds__(256) void infonce_wmma_kernel(
    const float* __restrict__ A,
    const float* __restrict__ P,
    const float* __restrict__ N,
    float* __restrict__ out,
    int B, float invB)
{
    constexpr int D = 512;
    __shared__ float wave_partial[8];

    const int lane = threadIdx.x & 31;
    const int wave = threadIdx.x >> 5;
    const int tile = blockIdx.x * 8 + wave;   // 16 rows per wave-tile
    const int r    = lane & 15;               // row-in-tile (also matrix M/N index)
    const int half = lane >> 4;               // upper lanes carry K = 2,3 (mod 4)
    const int koff = half * 2;

    float contrib = 0.0f;
    const int rowBase = tile * 16;

    if (rowBase < B) {                        // wave-uniform branch: EXEC stays all-ones
        const int  row   = rowBase + r;
        const bool valid = row < B;
        const size_t base = (size_t)(valid ? row : (B - 1)) * D + (size_t)koff;
        const float* ap = A + base;
        const float* pp = P + base;
        const float* np = N + base;

        v8f acc_ap = {};
        v8f acc_na = {};
        float na2 = 0.0f, np2 = 0.0f, nn2 = 0.0f;

        // Streamed once, no reuse: non-temporal loads keep the 192MB L2 clean.
        #pragma unroll 8
        for (int c = 0; c < D; c += 4) {
            v2f av = __builtin_nontemporal_load((const v2f*)(ap + c));
            v2f pv = __builtin_nontemporal_load((const v2f*)(pp + c));
            v2f nv = __builtin_nontemporal_load((const v2f*)(np + c));
            // acc_ap[m][n] += sum_k a[m][c+k] * p[n][c+k]   (diag = a.p)
            acc_ap = __builtin_amdgcn_wmma_f32_16x16x4_f32(
                false, av, false, pv, (short)0, acc_ap, false, false);
            // acc_na[m][n] += sum_k n[m][c+k] * a[n][c+k]   (diag = n.a)
            acc_na = __builtin_amdgcn_wmma_f32_16x16x4_f32(
                false, nv, false, av, (short)0, acc_na, false, false);
            // squared norms on VALU (co-executes with the matrix pipe)
            na2 = fmaf(av.y, av.y, fmaf(av.x, av.x, na2));
            np2 = fmaf(pv.y, pv.y, fmaf(pv.x, pv.x, np2));
            nn2 = fmaf(nv.y, nv.y, fmaf(nv.x, nv.x, nn2));
        }

        // combine K-half partial norms: row r total = lane r + lane r+16
        na2 += __shfl_xor(na2, 16, 32);
        np2 += __shfl_xor(np2, 16, 32);
        nn2 += __shfl_xor(nn2, 16, 32);

        // Diagonal extraction. D layout: VGPR j, lanes 0-15 -> (M=j, N=lane),
        // lanes 16-31 -> (M=8+j, N=lane-16). diag(m) lives in VGPR m&7 at lane
        // (m<8 ? m : m+16). Destination lane L (0..15) wants diag(L).
        float pos = 0.0f, neg = 0.0f;
        const int src = (r < 8) ? r : (r + 16);
        #pragma unroll
        for (int j = 0; j < 8; ++j) {
            float gp = __shfl(acc_ap[j], src, 32);
            float gn = __shfl(acc_na[j], src, 32);
            if ((r & 7) == j) { pos = gp; neg = gn; }
        }

        if (half == 0 && valid) {
            // F.normalize semantics: x / max(||x||, eps)
            const float norm_a = fmaxf(sqrtf(na2), NORM_EPS);
            const float norm_p = fmaxf(sqrtf(np2), NORM_EPS);
            const float norm_n = fmaxf(sqrtf(nn2), NORM_EPS);
            const float pos_sim = pos / (norm_a * norm_p);
            const float neg_sim = neg / (norm_n * norm_a);
            // -log_softmax(logits)[0] = softplus((l1 - l0)), numerically stable
            const float d = (neg_sim - pos_sim) * (1.0f / TEMPERATURE);
            const float loss = (d > 0.0f) ? (d + log1pf(expf(-d)))
                                          : log1pf(expf(d));
            contrib = loss * invB;
        }
    }

    // wave32 tree reduction
    #pragma unroll
    for (int off = 16; off > 0; off >>= 1)
        contrib += __shfl_xor(contrib, off, 32);
    if (lane == 0) wave_partial[wave] = contrib;
    __syncthreads();

    if (threadIdx.x == 0) {
        float s = 0.0f;
        #pragma unroll
        for (int w = 0; w < 8; ++w) s += wave_partial[w];
        atomicAdd(out, s);   // ~1K atomics total: negligible
    }
}

extern "C" void kernel_launch(void* const* d_in, const int* in_sizes, int n_in,
                              void* d_out, int out_size, void* d_ws, size_t ws_size,
                              hipStream_t stream) {
    const float* A = (const float*)d_in[0];   // anchors   [B, 512] fp32
    const float* P = (const float*)d_in[1];   // positives [B, 512] fp32
    const float* N = (const float*)d_in[2];   // negatives [B, 512] fp32
    float* out = (float*)d_out;               // scalar fp32 loss

    const int D = 512;
    const int B = in_sizes[0] / D;            // 131072
    const int tiles  = (B + 15) / 16;         // 16 rows per wave
    const int blocks = (tiles + 7) / 8;       // 8 waves (256 threads) per block

    infonce_zero_kernel<<<1, 1, 0, stream>>>(out);
    infonce_wmma_kernel<<<blocks, 256, 0, stream>>>(A, P, N, out, B, 1.0f / (float)B);
}